// RGCN_40209483826003
// MI455X (gfx1250) — compile-verified
//
#include <hip/hip_runtime.h>

// RGCN (2 layers) for MI455X / gfx1250, wave32, fp32 WMMA (V_WMMA_F32_16X16X4_F32).
//
// Strategy: edge-centric relation-sorted message passing. x (25.6MB) and the
// aggregation buffer (25.6MB) are L2-resident (192MB L2), so gather + atomic
// scatter traffic stays on-chip; GEMMs run on the WMMA pipe in exact fp32.

typedef float v2f __attribute__((ext_vector_type(2)));
typedef float v8f __attribute__((ext_vector_type(8)));

#define DFEAT 64
#define NREL  16

// meta[] int layout inside workspace:
//  [0..15]   counts per relation
//  [16..32]  offsets (exclusive prefix, 17 entries)
//  [64..80]  tileStart (prefix of per-relation 16-edge tile counts, 17 entries)
//  [96..111] scatter cursors
//  [112]     totalTiles
#define META_INTS 128

__device__ __forceinline__ v8f wmma_f32(v2f a, v2f b, v8f c) {
  // D = A(16x4 f32) * B(4x16 f32) + C(16x16 f32); exact fp32 semantics (RNE).
  return __builtin_amdgcn_wmma_f32_16x16x4_f32(
      /*neg_a=*/false, a, /*neg_b=*/false, b,
      /*c_mod=*/(short)0, c, /*reuse_a=*/false, /*reuse_b=*/false);
}

// ---------------- binning kernels ----------------

__global__ void meta_init_kernel(int* meta) {
  int i = threadIdx.x;
  if (i < META_INTS) meta[i] = 0;
}

__global__ void hist_kernel(const int* __restrict__ etype, int E, int* meta) {
  int e = blockIdx.x * blockDim.x + threadIdx.x;
  if (e < E) atomicAdd(&meta[etype[e] & (NREL - 1)], 1);
}

__global__ void scan_kernel(int* meta) {
  if (threadIdx.x != 0 || blockIdx.x != 0) return;
  int off = 0, toff = 0;
  meta[16] = 0;
  meta[64] = 0;
  for (int r = 0; r < NREL; ++r) {
    int c = meta[r];
    off += c;
    meta[16 + r + 1] = off;            // offsets
    toff += (c + 15) >> 4;             // tiles of 16 edges
    meta[64 + r + 1] = toff;           // tileStart
    meta[96 + r] = meta[16 + r];       // cursor = offset
  }
  meta[112] = toff;                    // totalTiles
}

__global__ void edge_sort_kernel(const int* __restrict__ edge_index,
                                 const int* __restrict__ etype, int E,
                                 int* meta,
                                 int* __restrict__ ssrc,
                                 int* __restrict__ sdst) {
  int e = blockIdx.x * blockDim.x + threadIdx.x;
  if (e >= E) return;
  int r = etype[e] & (NREL - 1);
  int pos = atomicAdd(&meta[96 + r], 1);
  ssrc[pos] = edge_index[e];           // row 0: src
  sdst[pos] = edge_index[E + e];       // row 1: dst
}

// ---------------- base: agg = x @ sw + b (+ x if residual) ----------------
// One wave computes a 16-row x 64-col output tile: 4 accumulators (v8f each),
// K=64 stepped 4 at a time -> 64 v_wmma_f32_16x16x4_f32 per wave tile.

__global__ void base_gemm_kernel(const float* __restrict__ xin,
                                 const float* __restrict__ sw,
                                 const float* __restrict__ bias,
                                 float* __restrict__ agg,
                                 int N, int residual) {
  const int lane = threadIdx.x & 31;
  const int wave = threadIdx.x >> 5;
  const int tile = blockIdx.x * (blockDim.x >> 5) + wave;
  const int ntiles = N >> 4;           // N = 100000 is a multiple of 16
  if (tile >= ntiles) return;

  const int m0    = tile << 4;
  const int l15   = lane & 15;
  const int khalf = (lane >> 4) << 1;  // lanes 0-15 -> K+0/1, lanes 16-31 -> K+2/3
  const int mrow  = m0 + l15;

  v8f acc[4];
#pragma unroll
  for (int nb = 0; nb < 4; ++nb) acc[nb] = (v8f)0.f;

#pragma unroll
  for (int k0 = 0; k0 < DFEAT; k0 += 4) {
    v2f a;
    a.x = xin[mrow * DFEAT + k0 + khalf + 0];
    a.y = xin[mrow * DFEAT + k0 + khalf + 1];
#pragma unroll
    for (int nb = 0; nb < 4; ++nb) {
      const int col = (nb << 4) + l15;
      v2f b;
      b.x = sw[(k0 + khalf + 0) * DFEAT + col];
      b.y = sw[(k0 + khalf + 1) * DFEAT + col];
      acc[nb] = wmma_f32(a, b, acc[nb]);
    }
  }

  // C/D layout: VGPR v, lanes 0-15 -> M = v, lanes 16-31 -> M = v + 8.
  const int mhi = (lane >> 4) << 3;
#pragma unroll
  for (int nb = 0; nb < 4; ++nb) {
    const int col = (nb << 4) + l15;
    const float bv = bias[col];
#pragma unroll
    for (int v = 0; v < 8; ++v) {
      const int row = m0 + mhi + v;
      float val = acc[nb][v] + bv;
      if (residual) val += xin[row * DFEAT + col];
      agg[row * DFEAT + col] = val;
    }
  }
}

// ---------------- edge messages: agg[dst] += x[src] @ W[etype] --------------
// Edges are sorted by relation; each wave owns one 16-edge tile of a single
// relation r, so B = W[r] is uniform. Gather 16 rows of x (L2-resident),
// 64 WMMAs, then global_atomic_add_f32 scatter into agg (L2-resident).

__global__ void edge_gemm_kernel(const float* __restrict__ xin,
                                 const float* __restrict__ W,
                                 const int* __restrict__ meta,
                                 const int* __restrict__ ssrc,
                                 const int* __restrict__ sdst,
                                 float* __restrict__ agg) {
  const int lane = threadIdx.x & 31;
  const int wave = threadIdx.x >> 5;
  const int tile = blockIdx.x * (blockDim.x >> 5) + wave;
  const int totalTiles = meta[112];
  if (tile >= totalTiles) return;

  // Find relation r with tileStart[r] <= tile < tileStart[r+1] (<=16 steps).
  int r = 0;
  while (tile >= meta[64 + r + 1]) ++r;

  const int tbase = meta[16 + r] + ((tile - meta[64 + r]) << 4);
  const int cend  = meta[16 + r + 1];

  const int l15   = lane & 15;
  const int khalf = (lane >> 4) << 1;
  const int e     = tbase + l15;
  const bool evalid = (e < cend);
  const int sidx  = evalid ? ssrc[e] : 0;
  const int didx  = evalid ? sdst[e] : 0;   // meaningful in lanes 0-15

  const float* __restrict__ Wr = W + (size_t)r * DFEAT * DFEAT;

  v8f acc[4];
#pragma unroll
  for (int nb = 0; nb < 4; ++nb) acc[nb] = (v8f)0.f;

#pragma unroll
  for (int k0 = 0; k0 < DFEAT; k0 += 4) {
    v2f a;
    a.x = evalid ? xin[sidx * DFEAT + k0 + khalf + 0] : 0.f;
    a.y = evalid ? xin[sidx * DFEAT + k0 + khalf + 1] : 0.f;
#pragma unroll
    for (int nb = 0; nb < 4; ++nb) {
      const int col = (nb << 4) + l15;
      v2f b;
      b.x = Wr[(k0 + khalf + 0) * DFEAT + col];
      b.y = Wr[(k0 + khalf + 1) * DFEAT + col];
      acc[nb] = wmma_f32(a, b, acc[nb]);
    }
  }

  const int mhi = (lane >> 4) << 3;
#pragma unroll
  for (int nb = 0; nb < 4; ++nb) {
    const int col = (nb << 4) + l15;
#pragma unroll
    for (int v = 0; v < 8; ++v) {
      const int m  = mhi + v;                 // edge slot 0..15 within tile
      const int de = __shfl(didx, m, 32);     // dst of that edge (wave32)
      if (tbase + m < cend)
        atomicAdd(&agg[(size_t)de * DFEAT + col], acc[nb][v]);
    }
  }
}

// ---------------- relu ----------------

__global__ void relu_kernel(const float4* __restrict__ in,
                            float4* __restrict__ out, int n4) {
  int i = blockIdx.x * blockDim.x + threadIdx.x;
  if (i >= n4) return;
  float4 v = in[i];
  v.x = fmaxf(v.x, 0.f); v.y = fmaxf(v.y, 0.f);
  v.z = fmaxf(v.z, 0.f); v.w = fmaxf(v.w, 0.f);
  out[i] = v;
}

// ---------------- host launch ----------------

extern "C" void kernel_launch(void* const* d_in, const int* in_sizes, int n_in,
                              void* d_out, int out_size, void* d_ws, size_t ws_size,
                              hipStream_t stream) {
  const float* x          = (const float*)d_in[0];
  const int*   edge_index = (const int*)d_in[1];
  const int*   edge_type  = (const int*)d_in[2];
  const float* W1         = (const float*)d_in[3];
  const float* sw1        = (const float*)d_in[4];
  const float* b1         = (const float*)d_in[5];
  const float* W2         = (const float*)d_in[6];
  const float* sw2        = (const float*)d_in[7];
  const float* b2         = (const float*)d_in[8];

  const int N = in_sizes[0] / DFEAT;
  const int E = in_sizes[2];

  // Workspace layout.
  char* ws = (char*)d_ws;
  int*   meta = (int*)ws;                                  // 512 B
  int*   ssrc = (int*)(ws + 512);                          // E ints
  int*   sdst = (int*)(ws + 512 + (size_t)E * 4);          // E ints
  float* agg  = (float*)(ws + 512 + (size_t)E * 8);        // N*64 floats
  float* h    = agg + (size_t)N * DFEAT;                   // N*64 floats

  // 1) bin edges by relation (shared by both layers)
  meta_init_kernel<<<1, META_INTS, 0, stream>>>(meta);
  hist_kernel<<<(E + 255) / 256, 256, 0, stream>>>(edge_type, E, meta);
  scan_kernel<<<1, 1, 0, stream>>>(meta);
  edge_sort_kernel<<<(E + 255) / 256, 256, 0, stream>>>(edge_index, edge_type, E,
                                                        meta, ssrc, sdst);

  const int ntilesN   = N / 16;                       // 6250 (N % 16 == 0)
  const int baseBlk   = (ntilesN + 3) / 4;            // 4 waves/block
  const int maxTiles  = E / 16 + NREL + 1;            // upper bound on totalTiles
  const int edgeBlk   = (maxTiles + 3) / 4;
  const int n4        = N * DFEAT / 4;

  // ---- layer 1 (residual = true) ----
  base_gemm_kernel<<<baseBlk, 128, 0, stream>>>(x, sw1, b1, agg, N, 1);
  edge_gemm_kernel<<<edgeBlk, 128, 0, stream>>>(x, W1, meta, ssrc, sdst, agg);
  relu_kernel<<<(n4 + 255) / 256, 256, 0, stream>>>((const float4*)agg,
                                                    (float4*)h, n4);

  // ---- layer 2 (no residual) ----
  base_gemm_kernel<<<baseBlk, 128, 0, stream>>>(h, sw2, b2, agg, N, 0);
  edge_gemm_kernel<<<edgeBlk, 128, 0, stream>>>(h, W2, meta, ssrc, sdst, agg);
  relu_kernel<<<(n4 + 255) / 256, 256, 0, stream>>>((const float4*)agg,
                                                    (float4*)d_out, n4);
}